// SSIMLoss_38869454029310
// MI455X (gfx1250) — compile-verified
//
#include <hip/hip_runtime.h>

// ---------------------------------------------------------------------------
// SSIM loss, MI455X (gfx1250, wave32).
// Box sums of the 5 maps (X, Y, XX, YY, XY) are computed as banded matmuls
// on the WMMA pipe: V_WMMA_F32_16X16X4_F32, D = A*B + C.
//   horizontal: S(16x16) = Tile(16x24) x Band(24x16)   (6 K=4 steps)
//   vertical:   D(16x16) = Band(16x24) x S(24x16)      (6 K=4 steps)
// Band[k][n] = 1 iff n <= k <= n+6  (7-wide window of ones).
// One wave per 16x16 output tile; 64 images * 32*32 tiles = 65536 tiles,
// 8 waves / 256-thread block -> 8192 blocks.
// ---------------------------------------------------------------------------

typedef float v2f __attribute__((ext_vector_type(2)));
typedef float v8f __attribute__((ext_vector_type(8)));

#define WMMA4(A, B, C) \
    __builtin_amdgcn_wmma_f32_16x16x4_f32(false, (A), false, (B), (short)0, (C), false, false)

#define IMG_W   512
#define IMG_PIX 262144       // 512*512
#define OUT_W   506
#define N_TILES 65536        // 64 images * 32*32 tiles
#define N_BLOCKS 8192        // N_TILES / 8 waves
#define TOTAL_VALID 16386304.0  // 64 * 506 * 506

__global__ __launch_bounds__(256) void ssim_tiles_kernel(
    const float* __restrict__ X, const float* __restrict__ Y,
    const float* __restrict__ data_range, float* __restrict__ block_sums)
{
    const int tid  = threadIdx.x;
    const int wave = tid >> 5;
    const int lane = tid & 31;

    const int tile = blockIdx.x * 8 + wave;          // 0 .. 65535
    const int n    = tile >> 10;                     // image index
    const int rem  = tile & 1023;
    const int y0   = (rem >> 5) << 4;                // tile origin (output coords)
    const int x0   = (rem & 31) << 4;

    const float* __restrict__ Xn = X + (size_t)n * IMG_PIX;
    const float* __restrict__ Yn = Y + (size_t)n * IMG_PIX;

    const int half = lane >> 4;                      // 0: lanes 0-15, 1: lanes 16-31
    const int l15  = lane & 15;

    // ---- horizontal pass: 5 maps x 2 rowsets, each accumulated over 6 K-steps
    v8f Sh[5][2];
#pragma unroll
    for (int m = 0; m < 5; ++m) {
        Sh[m][0] = (v8f){0.f,0.f,0.f,0.f,0.f,0.f,0.f,0.f};
        Sh[m][1] = (v8f){0.f,0.f,0.f,0.f,0.f,0.f,0.f,0.f};
    }

#pragma unroll
    for (int s = 0; s < 2; ++s) {
        // A-operand layout: lane holds row M = l15, K = 2*half + {0,1} (per K-step)
        const int row = min(y0 + s * 16 + l15, 511);        // clamp: only invalid
        const float* __restrict__ xr = Xn + (size_t)row * IMG_W;   // outputs touch
        const float* __restrict__ yr = Yn + (size_t)row * IMG_W;   // clamped data
#pragma unroll
        for (int j = 0; j < 6; ++j) {
            const int kb = 4 * j + 2 * half;                 // local k of this lane
            const int cb = min(x0 + kb, 510);                // keep v2f load in-bounds
            v2f ax = *(const v2f*)(xr + cb);
            v2f ay = *(const v2f*)(yr + cb);
            v2f A[5];
            A[0] = ax; A[1] = ay; A[2] = ax * ax; A[3] = ay * ay; A[4] = ax * ay;

            // B-operand (banded ones): VGPR0 holds rows kb, VGPR1 rows kb+1
            v2f B;
            B.x = (kb     >= l15 && kb     <= l15 + 6) ? 1.f : 0.f;
            B.y = (kb + 1 >= l15 && kb + 1 <= l15 + 6) ? 1.f : 0.f;

#pragma unroll
            for (int m = 0; m < 5; ++m)
                Sh[m][s] = WMMA4(A[m], B, Sh[m][s]);
        }
    }

    // ---- fetch S row r (r = 0..23, compile-time) at column l15, for all lanes.
    // C/D layout: tile t = r>>4, reg v = r&7, source half = (r>>3)&1.
    auto fetchRow = [&](const v8f S2[2], int r) -> float {
        float src = S2[r >> 4][r & 7];
        int srcLane = (((r >> 3) & 1) << 4) + l15;
        return __shfl(src, srcLane, 32);
    };

    // ---- vertical pass: D = Band(16x24) x S(24x16), per map
    v8f D[5];
#pragma unroll
    for (int m = 0; m < 5; ++m) {
        v8f acc = (v8f){0.f,0.f,0.f,0.f,0.f,0.f,0.f,0.f};
#pragma unroll
        for (int j = 0; j < 6; ++j) {
            const int kb = 4 * j + 2 * half;
            // A = band: lane row M = l15, A[M][k] = 1 iff M <= k <= M+6
            v2f Av;
            Av.x = (kb     >= l15 && kb     <= l15 + 6) ? 1.f : 0.f;
            Av.y = (kb + 1 >= l15 && kb + 1 <= l15 + 6) ? 1.f : 0.f;
            // B = S rows 4j..4j+3 in B layout: VGPR0 rows {4j, 4j+2}, VGPR1 {4j+1, 4j+3}
            float b0a = fetchRow(Sh[m], 4 * j + 0);
            float b0b = fetchRow(Sh[m], 4 * j + 2);
            float b1a = fetchRow(Sh[m], 4 * j + 1);
            float b1b = fetchRow(Sh[m], 4 * j + 3);
            v2f Bv;
            Bv.x = half ? b0b : b0a;
            Bv.y = half ? b1b : b1a;
            acc = WMMA4(Av, Bv, acc);
        }
        D[m] = acc;
    }

    // ---- pointwise SSIM + masked accumulation
    const float dr = data_range[n];
    float C1 = 0.01f * dr; C1 *= C1;
    float C2 = 0.03f * dr; C2 *= C2;
    const float inv49 = 1.0f / 49.0f;
    const float covn  = 49.0f / 48.0f;

    float partial = 0.f;
#pragma unroll
    for (int v = 0; v < 8; ++v) {
        float ux  = D[0][v] * inv49, uy  = D[1][v] * inv49;
        float uxx = D[2][v] * inv49, uyy = D[3][v] * inv49, uxy = D[4][v] * inv49;
        float vx  = covn * (uxx - ux * ux);
        float vy  = covn * (uyy - uy * uy);
        float vxy = covn * (uxy - ux * uy);
        float A1 = 2.f * ux * uy + C1;
        float A2 = 2.f * vxy + C2;
        float B1 = ux * ux + uy * uy + C1;
        float B2 = vx + vy + C2;
        float Sp = (A1 * A2) / (B1 * B2);
        int oy = y0 + v + 8 * half;          // C/D layout: reg v -> rows v, v+8
        int ox = x0 + l15;
        partial += (oy < OUT_W && ox < OUT_W) ? Sp : 0.f;
    }

    // wave reduce
#pragma unroll
    for (int off = 16; off >= 1; off >>= 1)
        partial += __shfl_xor(partial, off, 32);

    __shared__ float wsum[8];
    if (lane == 0) wsum[wave] = partial;
    __syncthreads();
    if (tid == 0) {
        float s = 0.f;
#pragma unroll
        for (int i = 0; i < 8; ++i) s += wsum[i];
        block_sums[blockIdx.x] = s;          // one deterministic slot per block
    }
}

__global__ __launch_bounds__(256) void ssim_final_kernel(
    const float* __restrict__ block_sums, float* __restrict__ out)
{
    double acc = 0.0;
    for (int i = threadIdx.x; i < N_BLOCKS; i += 256)
        acc += (double)block_sums[i];

    __shared__ double sdata[256];
    sdata[threadIdx.x] = acc;
    __syncthreads();
    for (int s = 128; s > 0; s >>= 1) {
        if (threadIdx.x < s) sdata[threadIdx.x] += sdata[threadIdx.x + s];
        __syncthreads();
    }
    if (threadIdx.x == 0)
        out[0] = 1.0f - (float)(sdata[0] / TOTAL_VALID);
}

extern "C" void kernel_launch(void* const* d_in, const int* in_sizes, int n_in,
                              void* d_out, int out_size, void* d_ws, size_t ws_size,
                              hipStream_t stream)
{
    const float* X  = (const float*)d_in[0];
    const float* Y  = (const float*)d_in[1];
    const float* dr = (const float*)d_in[2];
    float* block_sums = (float*)d_ws;        // needs 8192 * 4 B = 32 KB of d_ws

    ssim_tiles_kernel<<<N_BLOCKS, 256, 0, stream>>>(X, Y, dr, block_sums);
    ssim_final_kernel<<<1, 256, 0, stream>>>(block_sums, (float*)d_out);
}